// Detection_1640677507723
// MI455X (gfx1250) — compile-verified
//
#include <hip/hip_runtime.h>
#include <hip/hip_bf16.h>

// Problem constants (match reference)
#define NCLS   5
#define CM1    4        // foreground classes
#define TOPK   200
#define NB     128
#define NN     65536
#define THRESH 0.5f
#define OVR    0.5f

#define CAPMAX 6144     // per-(b,c) candidate capacity (expected ~3000)
#define KPAD   208      // 13 * 16 : top-K padded to WMMA tile multiple

typedef __attribute__((ext_vector_type(2))) float v2f;   // A/B of 16x16x4 f32 WMMA
typedef __attribute__((ext_vector_type(8))) float v8f;   // C/D 16x16 f32 accumulator

// LDS layout for select_nms_kernel (dynamic shared, 53248 bytes)
#define SM_KEYS   0        // 6144 x u64 = 49152 B ; reused as adj[208*208]=43264 B
#define SM_TOPS   49152    // float[208]
#define SM_TOPE   49984    // float[208]
#define SM_TOPV   50816    // float[208]
#define SM_RSLOT  51648    // int[208]
#define SM_RED    52480    // u64[8]
#define SM_KEEP   52544    // u8[208]
#define SM_SUPP   52752    // u8[208]
#define SM_BYTES  53248

__global__ void zero_counts_kernel(int* __restrict__ counts) {
  int i = blockIdx.x * blockDim.x + threadIdx.x;
  if (i < NB * CM1) counts[i] = 0;
}

// ---------------------------------------------------------------------------
// Kernel 1: fused softmax + decode + threshold + compaction (streaming pass).
// Softmax sums to 1 => at most one class > 0.5 per candidate => <=1 append.
// ---------------------------------------------------------------------------
__global__ void __launch_bounds__(256) compact_kernel(
    const float* __restrict__ loc,      // [B,N,2]
    const float* __restrict__ cls,      // [B,N,5]
    const float* __restrict__ defs,     // [N,2]
    int*   __restrict__ counts,         // [B*4]
    float4* __restrict__ cand,          // [B*4, cap] {score,start,end,_}
    int cap)
{
  int gid = blockIdx.x * 256 + threadIdx.x;     // 0 .. B*N-1
  int b = gid >> 16;
  int n = gid & (NN - 1);

  const float* cp = cls + (size_t)gid * NCLS;
  float x0 = cp[0], x1 = cp[1], x2 = cp[2], x3 = cp[3], x4 = cp[4];
  float m  = fmaxf(fmaxf(x0, x1), fmaxf(fmaxf(x2, x3), x4));
  float e0 = expf(x0 - m), e1 = expf(x1 - m), e2 = expf(x2 - m);
  float e3 = expf(x3 - m), e4 = expf(x4 - m);
  float s  = e0 + e1 + e2 + e3 + e4;

  float eb = fmaxf(fmaxf(e1, e2), fmaxf(e3, e4));   // best foreground
  float p  = eb / s;
  if (p > THRESH) {
    int c = (eb == e1) ? 0 : (eb == e2) ? 1 : (eb == e3) ? 2 : 3;
    // decode only for passing candidates (saves localization bandwidth)
    float l0 = loc[(size_t)gid * 2 + 0];
    float l1 = loc[(size_t)gid * 2 + 1];
    float dc = defs[n * 2 + 0];
    float dw = defs[n * 2 + 1];
    float ctr = dc + l0 * dw;
    float w   = dw * expf(l1);
    int bc = b * CM1 + c;
    int slot = atomicAdd(counts + bc, 1);
    if (slot < cap)
      cand[(size_t)bc * cap + slot] =
          make_float4(p, ctr - 0.5f * w, ctr + 0.5f * w, 0.f);
  }
}

// ---------------------------------------------------------------------------
// Kernel 2: per-(b,c) top-200 selection (iterative LDS argmax), WMMA-built
// IoU adjacency matrix, greedy NMS, output write.
// ---------------------------------------------------------------------------
__global__ void __launch_bounds__(256) select_nms_kernel(
    const int*    __restrict__ counts,
    const float4* __restrict__ cand,
    float*        __restrict__ out,     // [B,4,200,3]
    int cap)
{
  extern __shared__ unsigned char smem[];
  unsigned long long* keys = (unsigned long long*)(smem + SM_KEYS);
  unsigned char*      adj  = smem + SM_KEYS;          // reuse after selection
  float* top_s = (float*)(smem + SM_TOPS);
  float* top_e = (float*)(smem + SM_TOPE);
  float* top_v = (float*)(smem + SM_TOPV);
  int*   rslot = (int*)  (smem + SM_RSLOT);
  unsigned long long* red = (unsigned long long*)(smem + SM_RED);
  unsigned char* keep = smem + SM_KEEP;
  unsigned char* supp = smem + SM_SUPP;

  const int bc  = blockIdx.x;
  const int tid = threadIdx.x;
  const float4* cb = cand + (size_t)bc * cap;
  int M = counts[bc];
  if (M > cap) M = cap;

  // Phase A: build sort keys (score bits desc; slot for retrieval).
  for (int i = tid; i < M; i += 256) {
    float4 v = cb[i];                               // score in (0.5,1): positive
    keys[i] = ((unsigned long long)__float_as_uint(v.x) << 32) | (unsigned)i;
  }
  for (int i = tid; i < KPAD; i += 256) {
    top_s[i] = 0.f; top_e[i] = 0.f; top_v[i] = -1.f; rslot[i] = -1;
  }
  __syncthreads();

  // Phase B: 200 iterations of parallel argmax over LDS keys.
  for (int r = 0; r < TOPK; ++r) {
    unsigned long long mk = 0ULL;
    for (int i = tid; i < M; i += 256) {
      unsigned long long k = keys[i];
      if (k > mk) mk = k;
    }
    #pragma unroll
    for (int o = 16; o > 0; o >>= 1) {              // wave32 reduce
      unsigned long long t = __shfl_down(mk, (unsigned)o, 32);
      if (t > mk) mk = t;
    }
    if ((tid & 31) == 0) red[tid >> 5] = mk;
    __syncthreads();
    if (tid == 0) {
      unsigned long long best = red[0];
      #pragma unroll
      for (int wv = 1; wv < 8; ++wv) if (red[wv] > best) best = red[wv];
      if (best) {
        int sl = (int)(best & 0xFFFFFFFFULL);
        rslot[r] = sl;
        keys[sl] = 0ULL;                            // remove from pool
      }
    }
    __syncthreads();
  }

  // Gather selected candidates (bulk, off the critical argmax path).
  for (int r = tid; r < TOPK; r += 256) {
    int sl = rslot[r];
    if (sl >= 0) { float4 v = cb[sl]; top_v[r] = v.x; top_s[r] = v.y; top_e[r] = v.z; }
  }
  __syncthreads();   // keys now dead; region becomes adjacency matrix

  // Phase C: adjacency adj[i][j] = (iou > OVR) via V_WMMA_F32_16X16X4_F32.
  //   P  = len_i + len_j  (A row=[len,1,0,0], B rows=[1; len_j; 0; 0])
  //   De = e_i - e_j, Ds = s_i - s_j  (B second row negated)
  //   inter = max(0, 0.5*(P - |De| - |Ds|)); union = P - inter
  // A 16x4 f32 layout: VGPR0 = K0 (lanes0-15) / K2 (lanes16-31); VGPR1 = K1/K3.
  // B 4x16 mirrored. D: lanes0-15 -> M=v, lanes16-31 -> M=v+8; N = lane&15.
  {
    const int wave = tid >> 5, lane = tid & 31, h = lane >> 4, l = lane & 15;
    const float oneh = h ? 0.f : 1.f;
    for (int t = wave; t < 13 * 13; t += 8) {       // wave-uniform: EXEC all 1s
      int ti = t / 13, tj = t - ti * 13;
      float si = top_s[ti * 16 + l], ei = top_e[ti * 16 + l];
      float sj = top_s[tj * 16 + l], ej = top_e[tj * 16 + l];
      float li = ei - si, lj = ej - sj;
      v2f aP, bP, aE, bE, aS, bS;
      aP[0] = h ? 0.f : li;  aP[1] = oneh;
      bP[0] = oneh;          bP[1] = h ? 0.f : lj;
      aE[0] = h ? 0.f : ei;  aE[1] = oneh;
      bE[0] = oneh;          bE[1] = h ? 0.f : -ej;
      aS[0] = h ? 0.f : si;  aS[1] = oneh;
      bS[0] = oneh;          bS[1] = h ? 0.f : -sj;
      v8f z = {0.f, 0.f, 0.f, 0.f, 0.f, 0.f, 0.f, 0.f};
      v8f P  = __builtin_amdgcn_wmma_f32_16x16x4_f32(false, aP, false, bP,
                                                     (short)0, z, false, false);
      v8f De = __builtin_amdgcn_wmma_f32_16x16x4_f32(false, aE, false, bE,
                                                     (short)0, z, false, false);
      v8f Ds = __builtin_amdgcn_wmma_f32_16x16x4_f32(false, aS, false, bS,
                                                     (short)0, z, false, false);
      #pragma unroll
      for (int v = 0; v < 8; ++v) {
        float inter = 0.5f * (P[v] - fabsf(De[v]) - fabsf(Ds[v]));
        inter = fmaxf(inter, 0.f);
        float uni = P[v] - inter;                   // = len_i + len_j - inter
        int i = ti * 16 + h * 8 + v;
        int j = tj * 16 + l;
        adj[i * KPAD + j] = (inter > OVR * uni) ? (unsigned char)1
                                                : (unsigned char)0;
      }
    }
  }
  for (int i = tid; i < KPAD; i += 256) supp[i] = 0;
  __syncthreads();

  // Phase D: greedy NMS over sorted candidates.
  for (int i = 0; i < TOPK; ++i) {
    if (tid == 0)
      keep[i] = (top_v[i] > THRESH && !supp[i]) ? 1 : 0;
    __syncthreads();
    if (keep[i]) {
      const unsigned char* row = adj + i * KPAD;
      for (int j = i + 1 + tid; j < TOPK; j += 256) supp[j] |= row[j];
    }
    __syncthreads();
  }

  // Phase E: write (start,end,score) * keep; covers every output element.
  float* o = out + (size_t)bc * (TOPK * 3);
  for (int k = tid; k < TOPK; k += 256) {
    float kp = keep[k] ? 1.f : 0.f;
    o[k * 3 + 0] = top_s[k] * kp;
    o[k * 3 + 1] = top_e[k] * kp;
    o[k * 3 + 2] = top_v[k] * kp;
  }
}

extern "C" void kernel_launch(void* const* d_in, const int* in_sizes, int n_in,
                              void* d_out, int out_size, void* d_ws, size_t ws_size,
                              hipStream_t stream) {
  (void)in_sizes; (void)n_in; (void)out_size;
  const float* loc  = (const float*)d_in[0];   // [B,N,2]
  const float* cls  = (const float*)d_in[1];   // [B,N,5]
  const float* defs = (const float*)d_in[2];   // [N,2]
  float* out = (float*)d_out;                  // [B,4,200,3]

  // Workspace: counts (4KB-aligned header) + candidate arrays sized to fit ws.
  int* counts = (int*)d_ws;
  float4* cand = (float4*)((char*)d_ws + 4096);
  size_t avail = (ws_size > 4096) ? ws_size - 4096 : 0;
  int cap = (int)(avail / ((size_t)NB * CM1 * sizeof(float4)));
  if (cap > CAPMAX) cap = CAPMAX;
  if (cap < 1) cap = 1;

  zero_counts_kernel<<<(NB * CM1 + 255) / 256, 256, 0, stream>>>(counts);
  compact_kernel<<<(NB * NN) / 256, 256, 0, stream>>>(loc, cls, defs, counts,
                                                      cand, cap);
  select_nms_kernel<<<NB * CM1, 256, SM_BYTES, stream>>>(counts, cand, out, cap);
}